// AdvanceAttention_37151467110968
// MI455X (gfx1250) — compile-verified
//
#include <hip/hip_runtime.h>
#include <stdint.h>

// ---------------------------------------------------------------------------
// Problem constants (match reference)
// ---------------------------------------------------------------------------
#define BB   16
#define SS   16
#define DD   4096
#define HH   32
#define KVH  8
#define HDIM 128
#define MAXS 4096
#define NTILES 256     // T/16; tile 255 == the 16 fresh tokens
#define NQKV 6144      // H*HD + 2*KV*HD
#define NSPLIT 2       // T-dim splits for the flash kernel

typedef float v2f __attribute__((ext_vector_type(2)));
typedef float v8f __attribute__((ext_vector_type(8)));
typedef int   i32x4 __attribute__((ext_vector_type(4)));
typedef int   i32x8 __attribute__((ext_vector_type(8)));

#define WMMA_F32X4(a, b, c) \
  __builtin_amdgcn_wmma_f32_16x16x4_f32(false, (a), false, (b), (short)0, (c), false, false)

// ---------------------------------------------------------------------------
// CDNA5 async global->LDS (128b per lane) + asynccnt waits
// ---------------------------------------------------------------------------
static __device__ __forceinline__ void async_f4(const float* g, float* l) {
  asm volatile("global_load_async_to_lds_b128 %0, %1, off"
               :: "v"((unsigned)(uintptr_t)l),
                  "v"((unsigned long long)(uintptr_t)g)
               : "memory");
}

// wait until <= n async ops outstanding (in-order completion => oldest done)
#define ASYNC_WAIT(n) asm volatile("s_wait_asynccnt " #n ::: "memory")

// ---------------------------------------------------------------------------
// CDNA5 Tensor Data Mover: DMA a 16x128 f32 tile (row stride 1024 floats)
// from global into LDS, padding each 128-DWORD row with 4 DWORDs so rows land
// at the bank-conflict-free 132-float stride the WMMA fragment reads use.
// D# is fully uniform; issued by one wave, tracked on TENSORcnt.
// ---------------------------------------------------------------------------
static __device__ __forceinline__ void tdm_load_tile(const float* gsrc,
                                                     float* ldst) {
  const unsigned long long ga = (unsigned long long)(uintptr_t)gsrc;
  const unsigned lds = (unsigned)(uintptr_t)ldst;

  i32x4 g0;
  g0.x = 1;                                            // count=1, user D#
  g0.y = __builtin_amdgcn_readfirstlane((int)lds);     // lds_addr
  g0.z = __builtin_amdgcn_readfirstlane((int)(unsigned)ga);          // ga[31:0]
  g0.w = __builtin_amdgcn_readfirstlane(
      (int)((ga >> 32) & 0x1FFFFFFull) | (int)0x80000000u);  // ga[56:32]|type=2

  i32x8 g1;
  g1[0] = 0x20000            // data_size = 4B
        | (1 << 20)          // pad_enable
        | (6 << 22)          // pad_interval: every 128 DWORDs (one row)
        | (3 << 25);         // pad_amount: 4 DWORDs -> 132-float row stride
  g1[1] = 128 << 16;         // tensor_dim0 = 128 elements
  g1[2] = 16 << 16;          // tensor_dim1 = 16 lines
  g1[3] = 128 << 16;         // tile_dim0 = 128
  g1[4] = 16;                // tile_dim1 = 16, tile_dim2 = 0
  g1[5] = KVH * HDIM;        // tensor_dim0_stride = 1024 elements
  g1[6] = 0;
  g1[7] = 0;

  // 2D tensor: VADDR2/VADDR3 = NULL (2-group form)
  asm volatile("tensor_load_to_lds %0, %1" :: "s"(g0), "s"(g1) : "memory");
}

// wave32 xor-shuffle via ds_bpermute (stays inside 16-lane groups for m<16)
static __device__ __forceinline__ float xor_shfl(float x, int m) {
  int lane = (int)(threadIdx.x & 31);
  return __int_as_float(
      __builtin_amdgcn_ds_bpermute(((lane ^ m) << 2), __float_as_int(x)));
}

// ---------------------------------------------------------------------------
// GEMM core: block = 128 threads (4 waves) -> 16(M) x 64(N) tile.
// Double-buffered async staging; fragments batched 8 WMMA deep.
// ---------------------------------------------------------------------------
static __device__ __forceinline__ void stage_gemm_chunk(
    const float* __restrict__ A, const float* __restrict__ Bw, int lda,
    int ldb, int m0, int colBase, int k0, float* aBuf, float* bBuf, int tid) {
#pragma unroll
  for (int it = 0; it < 2; ++it) {   // A: 16x64 = 256 float4
    int e = it * 128 + tid, r = e >> 4, c4 = e & 15;
    async_f4(&A[(size_t)(m0 + r) * lda + k0 + c4 * 4], &aBuf[r * 68 + c4 * 4]);
  }
#pragma unroll
  for (int it = 0; it < 8; ++it) {   // B: 64x64 = 1024 float4
    int e = it * 128 + tid, r = e >> 4, c4 = e & 15;
    async_f4(&Bw[(size_t)(k0 + r) * ldb + colBase + c4 * 4],
             &bBuf[r * 68 + c4 * 4]);
  }
}

static __device__ __forceinline__ v8f gemm_tile(const float* __restrict__ A,
                                                const float* __restrict__ Bw,
                                                int lda, int ldb, int m0,
                                                int colBase, int K,
                                                float* aL, float* bL) {
  const int tid  = threadIdx.x;
  const int wave = tid >> 5;
  const int lane = tid & 31;
  const int hi   = lane >> 4;
  const int lo   = lane & 15;

  v8f acc = {0.f, 0.f, 0.f, 0.f, 0.f, 0.f, 0.f, 0.f};

  stage_gemm_chunk(A, Bw, lda, ldb, m0, colBase, 0, aL, bL, tid);

  for (int k0 = 0; k0 < K; k0 += 64) {
    const int cur = (k0 >> 6) & 1;
    float* aC = aL + cur * (16 * 68);
    float* bC = bL + cur * (64 * 68);
    if (k0 + 64 < K) {
      stage_gemm_chunk(A, Bw, lda, ldb, m0, colBase, k0 + 64,
                       aL + (cur ^ 1) * (16 * 68), bL + (cur ^ 1) * (64 * 68),
                       tid);
      ASYNC_WAIT(10);      // 10 ops/thread per chunk: oldest chunk complete
    } else {
      ASYNC_WAIT(0);
    }
    __syncthreads();

    const float* ar = &aC[lo * 68 + 2 * hi];              // A[m=lo][kk+2hi]
    const float* bc = &bC[2 * hi * 68 + wave * 16 + lo];  // B[kk+2hi][n=lo]
#pragma unroll
    for (int g8 = 0; g8 < 16; g8 += 8) {   // 16 k-steps in 2 batches of 8
      float2 af[8];
      float  bx[8], by[8];
#pragma unroll
      for (int u = 0; u < 8; ++u) {
        const int kk = (g8 + u) * 4;
        af[u] = *(const float2*)(ar + kk);
        bx[u] = bc[kk * 68];
        by[u] = bc[(kk + 1) * 68];
      }
#pragma unroll
      for (int u = 0; u < 8; ++u) {
        v2f a, b;
        a.x = af[u].x; a.y = af[u].y;
        b.x = bx[u];   b.y = by[u];
        acc = WMMA_F32X4(a, b, acc);
      }
    }
    __syncthreads();   // buffer consumed before next-next issue overwrites it
  }
  return acc;
}

// ---------------------------------------------------------------------------
// Kernel 1: fused QKV projection + RoPE  (grid = (96, 16), block = 128)
// ---------------------------------------------------------------------------
__global__ void __launch_bounds__(128)
qkv_rope_kernel(const float* __restrict__ x, const float* __restrict__ kin,
                const float* __restrict__ vin, const float* __restrict__ wq,
                const float* __restrict__ wk, const float* __restrict__ wv,
                const float* __restrict__ fcos, const float* __restrict__ fsin,
                float* __restrict__ xq, float* __restrict__ xk,
                float* __restrict__ xv) {
  __shared__ float aL[2 * 16 * 68];
  __shared__ float bL[2 * 64 * 68];

  const int nb = blockIdx.x * 64;   // column in concatenated [Q|K|V] space
  const int m0 = blockIdx.y * 16;   // row tile == batch b; rows are s = 0..15

  const float* A;
  const float* Bw;
  int ldb, col0, seg;
  if (nb < 4096)      { A = x;   Bw = wq; ldb = 4096; col0 = nb;        seg = 0; }
  else if (nb < 5120) { A = kin; Bw = wk; ldb = 1024; col0 = nb - 4096; seg = 1; }
  else                { A = vin; Bw = wv; ldb = 1024; col0 = nb - 5120; seg = 2; }

  v8f acc = gemm_tile(A, Bw, DD, ldb, m0, col0, DD, aL, bL);

  const int tid = threadIdx.x, wave = tid >> 5, lane = tid & 31;
  const int hi = lane >> 4, lo = lane & 15;
  const int ncol = col0 + wave * 16 + lo;

#pragma unroll
  for (int v = 0; v < 8; ++v) {
    const int s = v + 8 * hi;
    const int m = m0 + s;
    float val  = acc[v];
    float part = xor_shfl(val, 1);      // RoPE partner = adjacent lane
    if (seg != 2) {
      const int d  = ncol & 127;
      const float c  = fcos[s * 64 + (d >> 1)];
      const float sn = fsin[s * 64 + (d >> 1)];
      val = (d & 1) ? (part * sn + val * c) : (val * c - part * sn);
    }
    if (seg == 0)      xq[(size_t)m * 4096 + ncol] = val;
    else if (seg == 1) xk[(size_t)m * 1024 + ncol] = val;
    else               xv[(size_t)m * 1024 + ncol] = val;
  }
}

// ---------------------------------------------------------------------------
// Kernel 2: flash-attention partials over a T-range.
// grid = B*KV*NSPLIT = 256 blocks, block = 128 (4 waves = 4 heads per group).
// K/V tiles staged by the Tensor Data Mover (wave 0 issues, TENSORcnt-tracked,
// double buffered); Q fragments live in registers.
// ---------------------------------------------------------------------------
__global__ void __launch_bounds__(128)
attn_part_kernel(const float* __restrict__ xq, const float* __restrict__ xk,
                 const float* __restrict__ xv,
                 const float* __restrict__ cache_k,
                 const float* __restrict__ cache_v,
                 float* __restrict__ opart, float* __restrict__ mstat,
                 float* __restrict__ lstat) {
  __shared__ float kL[2 * 16 * 132];
  __shared__ float vL[2 * 16 * 132];
  __shared__ float pL[4][16 * 20];

  const int id = blockIdx.x;
  const int sp = id & 1;
  const int g  = (id >> 1) & 7;
  const int b  = id >> 4;
  const int tid = threadIdx.x, wave = tid >> 5, lane = tid & 31;
  const int hi = lane >> 4, lo = lane & 15;
  const int h = g * 4 + wave;

  const int tb = sp * (NTILES / NSPLIT);
  const int te = tb + (NTILES / NSPLIT);

  // Q fragments in registers: a.x/a.y = Q[lo][kk + 2*hi (+1)], kk = 4*t
  const size_t qbase = (((size_t)b * 16 + lo) * 32 + h) * 128 + 2 * hi;
  float2 qf[32];
#pragma unroll
  for (int t = 0; t < 32; ++t) qf[t] = *(const float2*)&xq[qbase + t * 4];

  v8f accO[8];
  float mI[8], lI[8];
#pragma unroll
  for (int nt = 0; nt < 8; ++nt)
    accO[nt] = (v8f){0.f, 0.f, 0.f, 0.f, 0.f, 0.f, 0.f, 0.f};
#pragma unroll
  for (int v = 0; v < 8; ++v) { mI[v] = -3.0e38f; lI[v] = 0.f; }

  const float scale = 0.08838834764831845f;  // 1/sqrt(128)

  // TDM staging: both cache ((b,t,g,:)) and fresh ws ((b,s,g,:)) tiles have
  // row stride KVH*HDIM = 1024 floats; only the base differs. Wave 0 issues
  // two DMA descriptors per tile (K and V).
  auto stage_kv = [&](int j, int bufi) {
    if (wave != 0) return;
    const int t0 = j * 16;
    const float* kS;
    const float* vS;
    if (j < NTILES - 1) {
      const size_t base = ((size_t)b * MAXS + t0) * (KVH * HDIM) + g * HDIM;
      kS = cache_k + base;  vS = cache_v + base;
    } else {
      const size_t base = (size_t)b * (16 * KVH * HDIM) + g * HDIM;
      kS = xk + base;       vS = xv + base;
    }
    tdm_load_tile(kS, kL + bufi * (16 * 132));
    tdm_load_tile(vS, vL + bufi * (16 * 132));
  };

  stage_kv(tb, 0);

  for (int j = tb; j < te; ++j) {
    const int cur = j & 1;
    if (j + 1 < te) {
      stage_kv(j + 1, cur ^ 1);                    // 2 TDM ops (wave 0)
      if (wave == 0) __builtin_amdgcn_s_wait_tensorcnt(2);  // oldest tile done
    } else {
      if (wave == 0) __builtin_amdgcn_s_wait_tensorcnt(0);
    }
    __syncthreads();

    const float* kC = kL + cur * (16 * 132);
    const float* vC = vL + cur * (16 * 132);

    // S = Q * K^T : 32 WMMA in 4 batches of 8
    v8f sA = {0.f, 0.f, 0.f, 0.f, 0.f, 0.f, 0.f, 0.f};
    const float* kr = &kC[lo * 132 + 2 * hi];
#pragma unroll
    for (int g8 = 0; g8 < 32; g8 += 8) {
      float2 bf[8];
#pragma unroll
      for (int u = 0; u < 8; ++u) bf[u] = *(const float2*)(kr + (g8 + u) * 4);
#pragma unroll
      for (int u = 0; u < 8; ++u) {
        v2f a, bv;
        a.x = qf[g8 + u].x;  a.y = qf[g8 + u].y;
        bv.x = bf[u].x;      bv.y = bf[u].y;
        sA = WMMA_F32X4(a, bv, sA);
      }
    }

    // online softmax update (row stats via 16-lane butterflies)
#pragma unroll
    for (int v = 0; v < 8; ++v) {
      float sv = sA[v] * scale;
      float mt = sv;
      mt = fmaxf(mt, xor_shfl(mt, 1));
      mt = fmaxf(mt, xor_shfl(mt, 2));
      mt = fmaxf(mt, xor_shfl(mt, 4));
      mt = fmaxf(mt, xor_shfl(mt, 8));
      const float mn   = fmaxf(mI[v], mt);
      const float corr = __expf(mI[v] - mn);
      const float pv   = __expf(sv - mn);
      float rs = pv;
      rs += xor_shfl(rs, 1);
      rs += xor_shfl(rs, 2);
      rs += xor_shfl(rs, 4);
      rs += xor_shfl(rs, 8);
      lI[v] = lI[v] * corr + rs;
      mI[v] = mn;
      pL[wave][(v + 8 * hi) * 20 + lo] = pv;   // P -> A-operand layout
#pragma unroll
      for (int nt = 0; nt < 8; ++nt) accO[nt][v] *= corr;
    }

    // O += P * V : 8 column tiles x 4 WMMA, fragments batched
    const float* pr = &pL[wave][lo * 20 + 2 * hi];
    float2 pf[4];
#pragma unroll
    for (int t = 0; t < 4; ++t) pf[t] = *(const float2*)(pr + t * 4);
#pragma unroll
    for (int nt = 0; nt < 8; ++nt) {
      float bx[4], by[4];
#pragma unroll
      for (int t = 0; t < 4; ++t) {
        const float* vb = &vC[(t * 4 + 2 * hi) * 132 + nt * 16 + lo];
        bx[t] = vb[0];
        by[t] = vb[132];
      }
      v8f acc = accO[nt];
#pragma unroll
      for (int t = 0; t < 4; ++t) {
        v2f a, bv;
        a.x = pf[t].x; a.y = pf[t].y;
        bv.x = bx[t];  bv.y = by[t];
        acc = WMMA_F32X4(a, bv, acc);
      }
      accO[nt] = acc;
    }
    __syncthreads();
  }

  // epilogue: unnormalized O + per-row stats for the combine pass
#pragma unroll
  for (int v = 0; v < 8; ++v) {
    const int s = v + 8 * hi;
    if (lo == 0) {
      const size_t si = (((size_t)sp * BB + b) * HH + h) * SS + s;
      mstat[si] = mI[v];
      lstat[si] = lI[v];
    }
#pragma unroll
    for (int nt = 0; nt < 8; ++nt) {
      opart[(size_t)sp * 1048576 +
            (((size_t)b * 16 + s) * 32 + h) * 128 + nt * 16 + lo] = accO[nt][v];
    }
  }
}

// ---------------------------------------------------------------------------
// Kernel 2b: combine the NSPLIT flash partials
// ---------------------------------------------------------------------------
__global__ void __launch_bounds__(256)
attn_combine_kernel(const float* __restrict__ opart,
                    const float* __restrict__ mstat,
                    const float* __restrict__ lstat,
                    float* __restrict__ attn) {
  const size_t idx = (size_t)blockIdx.x * 256 + threadIdx.x;  // < 1048576
  const int h = (int)((idx >> 7) & 31);
  const int s = (int)((idx >> 12) & 15);
  const int b = (int)(idx >> 16);
  const size_t s0 = (((size_t)0 * BB + b) * HH + h) * SS + s;
  const size_t s1 = (((size_t)1 * BB + b) * HH + h) * SS + s;
  const float m0 = mstat[s0], m1 = mstat[s1];
  const float M  = fmaxf(m0, m1);
  const float w0 = __expf(m0 - M), w1 = __expf(m1 - M);
  const float den = lstat[s0] * w0 + lstat[s1] * w1;
  attn[idx] = (opart[idx] * w0 + opart[1048576 + idx] * w1) / den;
}

// ---------------------------------------------------------------------------
// Kernel 3: output projection  attn(256x4096) @ wo(4096x4096) -> out
// ---------------------------------------------------------------------------
__global__ void __launch_bounds__(128)
oproj_kernel(const float* __restrict__ attn, const float* __restrict__ wo,
             float* __restrict__ out) {
  __shared__ float aL[2 * 16 * 68];
  __shared__ float bL[2 * 64 * 68];

  const int nb = blockIdx.x * 64;
  const int m0 = blockIdx.y * 16;
  v8f acc = gemm_tile(attn, wo, DD, DD, m0, nb, DD, aL, bL);

  const int tid = threadIdx.x, wave = tid >> 5, lane = tid & 31;
  const int hi = lane >> 4, lo = lane & 15;
#pragma unroll
  for (int v = 0; v < 8; ++v) {
    out[(size_t)(m0 + v + 8 * hi) * 4096 + nb + wave * 16 + lo] = acc[v];
  }
}

// ---------------------------------------------------------------------------
// Host launcher
// ---------------------------------------------------------------------------
extern "C" void kernel_launch(void* const* d_in, const int* in_sizes, int n_in,
                              void* d_out, int out_size, void* d_ws,
                              size_t ws_size, hipStream_t stream) {
  const float* x       = (const float*)d_in[0];
  const float* k       = (const float*)d_in[1];
  const float* v       = (const float*)d_in[2];
  const float* wq      = (const float*)d_in[3];
  const float* wk      = (const float*)d_in[4];
  const float* wv      = (const float*)d_in[5];
  const float* wo      = (const float*)d_in[6];
  const float* cache_k = (const float*)d_in[7];
  const float* cache_v = (const float*)d_in[8];
  const float* fcos    = (const float*)d_in[9];
  const float* fsin    = (const float*)d_in[10];
  float*       out     = (float*)d_out;

  // ws layout (floats): xq | xk | xv | attn | opart[2] | mstat | lstat (~18 MB)
  float* ws       = (float*)d_ws;
  float* ws_xq    = ws;                        // 1048576
  float* ws_xk    = ws_xq   + 1048576;         // 262144
  float* ws_xv    = ws_xk   + 262144;          // 262144
  float* ws_attn  = ws_xv   + 262144;          // 1048576
  float* ws_opart = ws_attn + 1048576;         // 2 * 1048576
  float* ws_m     = ws_opart + 2 * 1048576;    // 16384
  float* ws_l     = ws_m    + 16384;           // 16384

  qkv_rope_kernel<<<dim3(NQKV / 64, BB), 128, 0, stream>>>(
      x, k, v, wq, wk, wv, fcos, fsin, ws_xq, ws_xk, ws_xv);

  attn_part_kernel<<<dim3(BB * KVH * NSPLIT), 128, 0, stream>>>(
      ws_xq, ws_xk, ws_xv, cache_k, cache_v, ws_opart, ws_m, ws_l);

  attn_combine_kernel<<<dim3(4096), 256, 0, stream>>>(ws_opart, ws_m, ws_l,
                                                      ws_attn);

  oproj_kernel<<<dim3(DD / 64, BB), 128, 0, stream>>>(ws_attn, wo, out);
}